// DAGNN_77429670412905
// MI455X (gfx1250) — compile-verified
//
#include <hip/hip_runtime.h>
#include <cmath>

// ---------------------------------------------------------------------------
// DAGNN forward for MI455X (gfx1250, wave32).
//
// Strategy:
//  * 64 workgroups x 256 threads (8 waves). Each WG owns 32 batch rows
//    (one row per lane) and keeps its full activation slice in LDS:
//    act[32][PITCH] f32, PITCH=2049 dwords (odd => conflict-free gathers),
//    ~256KB -- uses CDNA5's 320KB-per-WGP LDS.
//  * Device-side levelization (setup kernel): level[i] = 1 + max level of
//    the 32 predecessors (~45 levels for this random DAG), counting-sorted
//    into `order` with per-level `offsets`. Main kernel: the 8 waves split
//    each level's nodes, s_barrier between levels => ~8x shorter critical
//    path than strict index order.
//  * Per node, the 32-row x 32-K dot product runs on V_WMMA_F32_16X16X4_F32
//    (full fp32): A = edge weights replicated across all 16 M rows
//    (documented A layout: V0={K0 lanes0-15, K2 lanes16-31}, V1={K1,K3});
//    B = gathered activations (N = lane&15; B assumed to mirror the C/D
//    striping: V0={K0,K2}, V1={K1,K3}). With all A rows equal, every row of
//    D is the desired dot vector, so extraction is one select of D[0].
//    Two accumulators cover batch rows 0-15 / 16-31; 8 K-chunks chain
//    through the C operand.
//  * x is staged global->LDS by the Tensor Data Mover (tensor_load_to_lds,
//    iterate mode: 32 iterations of one 512-dword row with
//    lds_addr_increment = PITCH), completed with s_wait_tensorcnt 0.
//    This toolchain (clang-23/therock) uses the 6-arg builtin form:
//    (v4u g0, v8i g1, v4i g2, v4i g3, v8i extra, i32 cpol).
// ---------------------------------------------------------------------------

#define N_NODES 2048
#define N_IN    512
#define N_OUT   512
#define DEG     32
#define BATCH   2048
#define ROWS    32                  // batch rows per workgroup (== wave width)
#define NWG     (BATCH / ROWS)      // 64 workgroups
#define PITCH   2049                // LDS row pitch (dwords), odd => no bank conflicts
#define THREADS 256
#define WAVES   (THREADS / 32)

typedef float        __attribute__((ext_vector_type(2))) v2f;
typedef float        __attribute__((ext_vector_type(8))) v8f;
typedef unsigned int __attribute__((ext_vector_type(4))) v4u;
typedef int          __attribute__((ext_vector_type(4))) v4i;
typedef int          __attribute__((ext_vector_type(8))) v8i;

// ---------------------------------------------------------------------------
// Setup: compute DAG levels, then counting-sort node ids by level.
// Single wave; LDS is in-order within a wave so the sequential dependency
// scan is safe. Deterministic (scatter is single-threaded).
// ---------------------------------------------------------------------------
__global__ __launch_bounds__(32) void dag_setup(const int* __restrict__ esrc,
                                                int* __restrict__ order,
                                                int* __restrict__ offsets,
                                                int* __restrict__ nlev) {
  __shared__ int lv[N_NODES];
  __shared__ int cnt[1544];
  __shared__ int maxl;
  const int lane = threadIdx.x;

  for (int i = lane; i < N_IN; i += 32) lv[i] = 0;
  for (int i = lane; i < 1544; i += 32) cnt[i] = 0;
  if (lane == 0) maxl = 0;
  __syncthreads();

  for (int i = N_IN; i < N_NODES; ++i) {
    int s = esrc[(i - N_IN) * DEG + lane];   // s < i, already resolved
    int l = lv[s];
    for (int o = 16; o; o >>= 1) {
      int t = __shfl_down(l, o, 32);
      l = (t > l) ? t : l;
    }
    if (lane == 0) lv[i] = l + 1;
  }
  __syncthreads();

  for (int i = N_IN + lane; i < N_NODES; i += 32) {
    atomicAdd(&cnt[lv[i]], 1);
    atomicMax(&maxl, lv[i]);
  }
  __syncthreads();

  if (lane == 0) {
    int NL = maxl;
    int run = 0;
    for (int l = 1; l <= NL; ++l) {
      offsets[l] = run;
      run += cnt[l];
      cnt[l] = offsets[l];                   // reuse as scatter cursor
    }
    offsets[NL + 1] = run;                   // == 1536
    for (int i = N_IN; i < N_NODES; ++i) {
      int l = lv[i];
      order[cnt[l]++] = i;
    }
    *nlev = NL;
  }
}

// ---------------------------------------------------------------------------
// Main: level-parallel DAG evaluation with WMMA dot products.
// ---------------------------------------------------------------------------
__global__ __launch_bounds__(THREADS) void dag_main(const float* __restrict__ x,
                                                    const float* __restrict__ ew,
                                                    const float* __restrict__ bias,
                                                    const int*   __restrict__ esrc,
                                                    const int*   __restrict__ order,
                                                    const int*   __restrict__ offsets,
                                                    const int*   __restrict__ nlev,
                                                    float*       __restrict__ out) {
  extern __shared__ float act[];             // [ROWS][PITCH]
  const int tid  = threadIdx.x;
  const int lane = tid & 31;
  const int wave = tid >> 5;
  const int wg   = blockIdx.x;

  // ---- Stage x rows into LDS via the Tensor Data Mover (async DMA) ----
#if defined(__has_builtin) && __has_builtin(__builtin_amdgcn_tensor_load_to_lds)
  if (wave == 0) {
    unsigned long long ga =
        (unsigned long long)(const void*)(x + (size_t)wg * ROWS * N_IN);
    v4u g0;
    g0[0] = 1u;                                            // count=1, user desc
    g0[1] = 0u;                                            // lds_addr = 0 bytes
    g0[2] = (unsigned)(ga & 0xFFFFFFFFu);                  // global_addr[31:0]
    g0[3] = (unsigned)((ga >> 32) & 0x01FFFFFFu) | (2u << 30); // addr[56:32], type=2
    v8i g1;
    g1[0] = (2 << 16) | (1 << 19);                         // data_size=4B, iterate_enable
    g1[1] = (N_IN & 0xFFFF) << 16;                         // tensor_dim0[15:0] @ bits63:48
    g1[2] = ((N_IN >> 16) & 0xFFFF) | (ROWS << 16);        // dim0 hi | tensor_dim1 lo
    g1[3] = (N_IN << 16);                                  // dim1 hi(0) | tile_dim0=512
    g1[4] = 1;                                             // tile_dim1=1, tile_dim2=0
    g1[5] = N_IN;                                          // tensor_dim0_stride lo32
    g1[6] = 0;
    g1[7] = 0;                                             // dim1_stride (ignored)
    v4i g2;
    g2[0] = 0;                                             // tensor_dim2
    g2[1] = PITCH;                                         // lds_addr_increment (elems)
    g2[2] = N_IN;                                          // global_addr_increment lo32
    g2[3] = ((ROWS - 1) << 16);                            // iterate_count = 31 -> 32 iters
    v4i g3 = {0, 0, 0, 0};
    v8i gx = {0, 0, 0, 0, 0, 0, 0, 0};                     // extra group (clang-23 form)
    __builtin_amdgcn_tensor_load_to_lds(g0, g1, g2, g3, gx, 0);
    __builtin_amdgcn_s_wait_tensorcnt(0);
  }
#else
  for (int r = 0; r < ROWS; ++r)
    for (int n = tid; n < N_IN; n += THREADS)
      act[r * PITCH + n] = x[((size_t)wg * ROWS + r) * N_IN + n];
#endif
  __syncthreads();

  const int NL   = nlev[0];
  const int half = lane >> 4;                // A/B hi-half holds K2,K3
  const int r15  = lane & 15;                // N index (batch row within group)

  for (int L = 1; L <= NL; ++L) {
    const int beg = offsets[L];
    const int end = offsets[L + 1];
    for (int j = beg + wave; j < end; j += WAVES) {
      const int node = order[j];
      const int eb   = (node - N_IN) * DEG;
      v8f acc1 = {0.f, 0.f, 0.f, 0.f, 0.f, 0.f, 0.f, 0.f};  // batch rows 0-15
      v8f acc2 = {0.f, 0.f, 0.f, 0.f, 0.f, 0.f, 0.f, 0.f};  // batch rows 16-31
#pragma unroll
      for (int c = 0; c < DEG / 4; ++c) {
        const int k0 = eb + 4 * c + (half << 1);     // lanes0-15: K0,K1; lanes16-31: K2,K3
        v2f A;
        A[0] = ew[k0];
        A[1] = ew[k0 + 1];                           // weights, same for every M row
        const int s0 = esrc[k0];
        const int s1 = esrc[k0 + 1];
        v2f B1, B2;
        B1[0] = act[r15 * PITCH + s0];               // conflict-free: bank = r15 + s
        B1[1] = act[r15 * PITCH + s1];
        B2[0] = act[(r15 + 16) * PITCH + s0];
        B2[1] = act[(r15 + 16) * PITCH + s1];
        acc1 = __builtin_amdgcn_wmma_f32_16x16x4_f32(false, A, false, B1,
                                                     (short)0, acc1, false, false);
        acc2 = __builtin_amdgcn_wmma_f32_16x16x4_f32(false, A, false, B2,
                                                     (short)0, acc2, false, false);
      }
      // Every D row equals the dot vector; D VGPR0 lane n holds pre[group_row n].
      const float pre = (half ? acc2[0] : acc1[0]) + bias[node];
      act[lane * PITCH + node] = tanhf(pre);         // bank = lane + node: conflict-free
    }
    __syncthreads();
  }

  // Emit last N_OUT columns, coalesced per row.
  for (int r = 0; r < ROWS; ++r)
    for (int o = tid; o < N_OUT; o += THREADS)
      out[((size_t)wg * ROWS + r) * N_OUT + o] =
          act[r * PITCH + (N_NODES - N_OUT) + o];
}

// ---------------------------------------------------------------------------
extern "C" void kernel_launch(void* const* d_in, const int* in_sizes, int n_in,
                              void* d_out, int out_size, void* d_ws, size_t ws_size,
                              hipStream_t stream) {
  const float* x    = (const float*)d_in[0];
  const float* ew   = (const float*)d_in[1];
  const float* bias = (const float*)d_in[2];
  const int*   esrc = (const int*)d_in[3];
  // d_in[4] = edge_dst (implicit via layout), d_in[5] = n_out (== 512)
  float* out = (float*)d_out;

  int* ws      = (int*)d_ws;
  int* order   = ws;                 // 2048 ints (1536 used)
  int* offsets = ws + 2048;          // 2052 ints
  int* nlev    = ws + 2048 + 2052;   // 1 int

  const size_t shmem = (size_t)ROWS * PITCH * sizeof(float);  // ~256KB
  (void)hipFuncSetAttribute((const void*)dag_main,
                            hipFuncAttributeMaxDynamicSharedMemorySize,
                            (int)shmem);

  hipLaunchKernelGGL(dag_setup, dim3(1), dim3(32), 0, stream,
                     esrc, order, offsets, nlev);
  hipLaunchKernelGGL(dag_main, dim3(NWG), dim3(THREADS), shmem, stream,
                     x, ew, bias, esrc, order, offsets, nlev, out);
}